// RaggedHSTUAttn_34067680592298
// MI455X (gfx1250) — compile-verified
//
#include <hip/hip_runtime.h>
#include <hip/hip_bf16.h>

typedef __attribute__((ext_vector_type(2))) float v2f;
typedef __attribute__((ext_vector_type(4))) float v4f;
typedef __attribute__((ext_vector_type(8))) float v8f;
typedef int v4i __attribute__((vector_size(16)));

#define ZSEQ   128
#define NHEAD  4
#define DQ     128
#define ALPHA  0.08838834764831843f
#define INV_N  (1.0f / 512.0f)

// LDS layout (in floats). Pitch 132 (=128+4) makes the stride-QPITCH fragment
// reads hit 16 distinct banks (132 mod 64 = 4). Pitch 68 for P likewise.
#define QPITCH 132
#define PPITCH 68
#define KS_OFF (64 * QPITCH)
#define VS_OFF (2 * 64 * QPITCH)
#define PS_OFF (3 * 64 * QPITCH)
#define SMEM_FLOATS (3 * 64 * QPITCH + 4 * 16 * PPITCH)   // 29,696 floats = 118,784 B

// ---- CDNA5 async global->LDS path (ASYNCcnt), guarded for toolchain support ----
#if defined(__has_builtin)
#  if __has_builtin(__builtin_amdgcn_global_load_async_to_lds_b128) && \
      __has_builtin(__builtin_amdgcn_s_wait_asynccnt)
#    define USE_ASYNC_LDS 1
#  endif
#endif
#ifndef USE_ASYNC_LDS
#  define USE_ASYNC_LDS 0
#endif

// Builtin parameter types (from clang diagnostic): global int4* src, local int4* dst.
typedef __attribute__((address_space(1))) v4i gbl_v4i;
typedef __attribute__((address_space(3))) v4i lds_v4i;

// Flat LDS pointer -> raw LDS offset: per CDNA5 ISA §10.2 the hardware uses
// addr[31:0] of a shared-aperture flat address directly as the LDS address.
__device__ __forceinline__ lds_v4i* to_lds(const void* p) {
    return (lds_v4i*)(uint32_t)(uintptr_t)p;
}
__device__ __forceinline__ gbl_v4i* to_gbl(const void* p) {
    return (gbl_v4i*)(uintptr_t)p;
}

__global__ __launch_bounds__(128)
void hstu_ragged_attn_kernel(const float* __restrict__ qkv,
                             const int*   __restrict__ seq_offsets,
                             float*       __restrict__ out)
{
    extern __shared__ float smem[];
    float* Qs = smem;
    float* Ks = smem + KS_OFF;
    float* Vs = smem + VS_OFF;
    float* Ps = smem + PS_OFF;

    const int bid = blockIdx.x;
    const int t   = bid & 7;          // 64-row tile index
    const int h   = (bid >> 3) & 3;   // head
    const int b   = bid >> 5;         // sequence

    const int off = seq_offsets[b];
    const int n   = seq_offsets[b + 1] - off;
    const int row0 = t * 64;
    if (row0 >= n) return;            // block-uniform

    const int tid  = threadIdx.x;
    const int w    = tid >> 5;        // wave id (0..3), owns rows w*16..w*16+15
    const int lane = tid & 31;
    const int half = lane >> 4;       // 0: K even pair, 1: K odd pair (ISA A/B layout)
    const int lr   = lane & 15;

    // ---------------- load Q tile (64 rows x 128) into LDS ----------------
    // Ragged guard by clamping the token index: clamped rows load real, finite
    // data; the score mask (gr < n) zeroes their effect and their output rows
    // are never stored. Keeps EXEC all-ones (WMMA requirement) and lets every
    // lane issue an async copy.
    for (int i = tid; i < 64 * 32; i += 128) {
        const int r  = i >> 5;
        const int c4 = (i & 31) << 2;
        const int gr = min(row0 + r, n - 1);
        const float* gp = qkv + (size_t)(off + gr) * (NHEAD * 3 * DQ) + h * (3 * DQ) + c4;
#if USE_ASYNC_LDS
        __builtin_amdgcn_global_load_async_to_lds_b128(
            to_gbl(gp), to_lds(&Qs[r * QPITCH + c4]), 0, 0);
#else
        *(v4f*)&Qs[r * QPITCH + c4] = *(const v4f*)gp;
#endif
    }
#if USE_ASYNC_LDS
    __builtin_amdgcn_s_wait_asynccnt(0);
#endif
    __syncthreads();

    v8f oacc[8];
#pragma unroll
    for (int d = 0; d < 8; ++d) oacc[d] = (v8f){0.f,0.f,0.f,0.f,0.f,0.f,0.f,0.f};

    float* Pw = Ps + w * 16 * PPITCH;                 // per-wave P staging (16x64)
    const int abase = (w * 16 + lr) * QPITCH + 2 * half;
    const int bbase = lr * QPITCH + 2 * half;

    // causal: only column groups with col0 <= last row of this tile (and < n)
    const int ncg = (min(row0 + 64, n) + 63) >> 6;

    for (int cg = 0; cg < ncg; ++cg) {
        const int col0 = cg * 64;

        __syncthreads();              // previous group's K/V reads done
        // -------- load K and V tiles (64 cols x 128) into LDS --------
        for (int i = tid; i < 64 * 32; i += 128) {
            const int r  = i >> 5;
            const int c4 = (i & 31) << 2;
            const int gc = min(col0 + r, n - 1);
            const float* gp = qkv + (size_t)(off + gc) * (NHEAD * 3 * DQ) + h * (3 * DQ) + DQ + c4;
#if USE_ASYNC_LDS
            __builtin_amdgcn_global_load_async_to_lds_b128(
                to_gbl(gp), to_lds(&Ks[r * QPITCH + c4]), 0, 0);
            __builtin_amdgcn_global_load_async_to_lds_b128(
                to_gbl(gp + DQ), to_lds(&Vs[r * QPITCH + c4]), 0, 0);
#else
            *(v4f*)&Ks[r * QPITCH + c4] = *(const v4f*)gp;
            *(v4f*)&Vs[r * QPITCH + c4] = *(const v4f*)(gp + DQ);
#endif
        }
#if USE_ASYNC_LDS
        __builtin_amdgcn_s_wait_asynccnt(0);
#endif
        __syncthreads();

        // -------- S = Q * K^T  (16x64 strip per wave) --------
        v8f acc[4];
#pragma unroll
        for (int ct = 0; ct < 4; ++ct) acc[ct] = (v8f){0.f,0.f,0.f,0.f,0.f,0.f,0.f,0.f};

        for (int kk = 0; kk < DQ; kk += 4) {
            v2f a;
            a.x = Qs[abase + kk];
            a.y = Qs[abase + kk + 1];
#pragma unroll
            for (int ct = 0; ct < 4; ++ct) {
                v2f bv;
                const int kb = bbase + ct * 16 * QPITCH + kk;
                bv.x = Ks[kb];
                bv.y = Ks[kb + 1];
                acc[ct] = __builtin_amdgcn_wmma_f32_16x16x4_f32(
                    false, a, false, bv, (short)0, acc[ct], false, false);
            }
        }

        // -------- P = silu(alpha*S)/512, mask, stage to LDS for A-layout reads --------
#pragma unroll
        for (int ct = 0; ct < 4; ++ct) {
#pragma unroll
            for (int r = 0; r < 8; ++r) {
                const float x = acc[ct][r] * ALPHA;
                const float p = x * (1.0f / (1.0f + __expf(-x))) * INV_N;
                const int gr = row0 + w * 16 + r + 8 * half;   // D-layout: M = r + 8*(lane/16)
                const int gc = col0 + ct * 16 + lr;            // D-layout: N = lane%16
                const bool ok = (gc <= gr) && (gc < n) && (gr < n);
                Pw[(r + 8 * half) * PPITCH + ct * 16 + lr] = ok ? p : 0.0f;
            }
        }
        // wave-private LDS region: same-wave DS ops are in-order, no barrier needed

        // -------- O += P * V  (16x128 strip per wave) --------
        for (int kk = 0; kk < 64; kk += 4) {
            v2f a;
            const int pb = lr * PPITCH + kk + 2 * half;
            a.x = Pw[pb];
            a.y = Pw[pb + 1];
#pragma unroll
            for (int d = 0; d < 8; ++d) {
                v2f bv;
                const int vb = (kk + 2 * half) * QPITCH + d * 16 + lr;
                bv.x = Vs[vb];
                bv.y = Vs[vb + QPITCH];
                oacc[d] = __builtin_amdgcn_wmma_f32_16x16x4_f32(
                    false, a, false, bv, (short)0, oacc[d], false, false);
            }
        }
    }

    // ---------------- store O (only valid rows) ----------------
#pragma unroll
    for (int d = 0; d < 8; ++d) {
#pragma unroll
        for (int r = 0; r < 8; ++r) {
            const int gr = row0 + w * 16 + r + 8 * half;
            if (gr < n) {
                out[(size_t)(off + gr) * (NHEAD * DQ) + h * DQ + d * 16 + lr] = oacc[d][r];
            }
        }
    }
}

extern "C" void kernel_launch(void* const* d_in, const int* in_sizes, int n_in,
                              void* d_out, int out_size, void* d_ws, size_t ws_size,
                              hipStream_t stream) {
    (void)in_sizes; (void)n_in; (void)out_size; (void)d_ws; (void)ws_size;
    const float* qkv         = (const float*)d_in[0];
    const int*   seq_offsets = (const int*)d_in[1];
    // d_in[2] = timestamps: unused by the reference math
    float* out = (float*)d_out;

    const size_t shmem = (size_t)SMEM_FLOATS * sizeof(float);   // 118,784 B (< 320 KB WGP LDS)
    (void)hipFuncSetAttribute(reinterpret_cast<const void*>(hstu_ragged_attn_kernel),
                              hipFuncAttributeMaxDynamicSharedMemorySize, (int)shmem);

    const int blocks = ZSEQ * NHEAD * 8;   // (seq, head, 64-row tile)
    hipLaunchKernelGGL(hstu_ragged_attn_kernel, dim3(blocks), dim3(128), shmem, stream,
                       qkv, seq_offsets, out);
}